// AttnDecoderRNN2_40767829574448
// MI455X (gfx1250) — compile-verified
//
#include <hip/hip_runtime.h>
#include <hip/hip_bf16.h>
#include <math.h>

// ---------------------------------------------------------------------------
// AttnDecoderRNN2 single step, N=64, T=2048, E=512, A=256, D=512, S=64, O=80
// Algorithmic reductions (exact): attention mask covers only t=0..min(9,L-1);
// softmax global-max shift cancels; LSTM h=c=0 so Whh* are dead.
// Heavy work = (64x1088)@(1088x8192) and (64x2048)@(2048x8192) fp32 GEMMs,
// done with V_WMMA_F32_16X16X4_F32 (CDNA5 native fp32 matrix op).
// GEMM waves process 2 N-tiles each so every A v2f load feeds 2 WMMAs.
// ---------------------------------------------------------------------------

typedef float v2f __attribute__((ext_vector_type(2)));
typedef float v8f __attribute__((ext_vector_type(8)));

#define NB   64
#define TENC 2048
#define EDIM 512
#define ADIM 256
#define DDIM 512
#define SDIM 64
#define ODIM 80
#define H4   2048
#define G4   (4*H4)           // 8192
#define KIN0 (EDIM+DDIM+SDIM) // 1088
#define DECD (H4+EDIM)        // 2560
#define NOUT 160              // O*R_FAC

__device__ __forceinline__ float sigmoidf(float x) { return 1.0f / (1.0f + __expf(-x)); }

// ---------------------------------------------------------------------------
// sp = tanh(relu(speed*sp_W1 + sp_b1) @ sp_W2 + sp_b2)   (per batch, 512-d)
// ---------------------------------------------------------------------------
__global__ __launch_bounds__(256) void sp_kernel(
    const float* __restrict__ speed,
    const float* __restrict__ sp_W1, const float* __restrict__ sp_b1,
    const float* __restrict__ sp_W2, const float* __restrict__ sp_b2,
    float* __restrict__ sp_out)
{
    const int n = blockIdx.x, tid = threadIdx.x;
    __shared__ float s_h[DDIM];
    const float spd = speed[n];
    for (int k = tid; k < DDIM; k += 256)
        s_h[k] = fmaxf(spd * sp_W1[k] + sp_b1[k], 0.0f);
    __syncthreads();
    for (int d = tid; d < EDIM; d += 256) {
        float acc = sp_b2[d];
        for (int k = 0; k < DDIM; ++k)
            acc += s_h[k] * sp_W2[k * EDIM + d];   // coalesced in d
        sp_out[n * EDIM + d] = tanhf(acc);
    }
}

// ---------------------------------------------------------------------------
// prenet: x = [input_dec(80), spkr(64)] ; p = relu(relu(x@W1+b1)@W2+b2)
// writes p into in_lstm[n, 0:512]
// ---------------------------------------------------------------------------
__global__ __launch_bounds__(256) void prenet_kernel(
    const float* __restrict__ input_dec, const float* __restrict__ spkr_vec,
    const float* __restrict__ pre_W1, const float* __restrict__ pre_b1,
    const float* __restrict__ pre_W2, const float* __restrict__ pre_b2,
    float* __restrict__ in_lstm)
{
    const int n = blockIdx.x, tid = threadIdx.x;
    __shared__ float s_x[ODIM + SDIM];   // 144
    __shared__ float s_h[2 * DDIM];      // 1024
    if (tid < ODIM)                 s_x[tid] = input_dec[n * ODIM + tid];
    else if (tid < ODIM + SDIM)     s_x[tid] = spkr_vec[n * SDIM + (tid - ODIM)];
    __syncthreads();
    for (int j = tid; j < 2 * DDIM; j += 256) {
        float acc = pre_b1[j];
        for (int k = 0; k < ODIM + SDIM; ++k)
            acc += s_x[k] * pre_W1[k * (2 * DDIM) + j];
        s_h[j] = fmaxf(acc, 0.0f);
    }
    __syncthreads();
    for (int d = tid; d < DDIM; d += 256) {
        float acc = pre_b2[d];
        for (int j = 0; j < 2 * DDIM; ++j)
            acc += s_h[j] * pre_W2[j * DDIM + d];
        in_lstm[n * KIN0 + d] = fmaxf(acc, 0.0f);
    }
}

// ---------------------------------------------------------------------------
// Attention (masked positions t = 0..min(9, L-1) only) + context.
// Writes context into in_lstm[:,512:1024], dec_ws[:,2048:2560],
// d_out context region, d_out dec_output tail; copies spkr into in_lstm tail.
// ---------------------------------------------------------------------------
__global__ __launch_bounds__(256) void attn_kernel(
    const float* __restrict__ input_enc, const float* __restrict__ spkr_vec,
    const float* __restrict__ speed, const int* __restrict__ lengths_enc,
    const float* __restrict__ We, const float* __restrict__ be,
    const float* __restrict__ Ws, const float* __restrict__ conv_w,
    const float* __restrict__ w_proj, const float* __restrict__ b_proj,
    const float* __restrict__ W_speed, const float* __restrict__ sp_ws,
    float* __restrict__ in_lstm, float* __restrict__ dec_ws,
    float* __restrict__ out_ctx, float* __restrict__ out_dec)
{
    const int n = blockIdx.x, a = threadIdx.x;   // a = attention channel 0..255
    __shared__ float s_bspkr[ADIM];
    __shared__ float s_enc[EDIM];
    __shared__ float s_red[256];
    __shared__ float s_logit[10];
    __shared__ float s_att[10];

    const float spd = speed[n];
    const int   L1  = lengths_enc[n] - 1;
    const int   tmax = (L1 < 9) ? L1 : 9;        // masked positions 0..tmax

    // bias_spkr[a] = softsign(spkr . Ws[:,a])
    const float* sv = spkr_vec + n * SDIM;
    float acc = 0.0f;
    for (int e = 0; e < SDIM; ++e) acc += sv[e] * Ws[e * ADIM + a];
    s_bspkr[a] = acc / (1.0f + fabsf(acc));
    __syncthreads();

    const float wpa  = w_proj[a];
    const float bspd = spd * W_speed[a];

    for (int t = 0; t <= tmax; ++t) {
        const float* er = input_enc + ((size_t)n * TENC + t) * EDIM;
        for (int e = a; e < EDIM; e += 256) s_enc[e] = er[e];
        __syncthreads();
        float d = be[a];
        for (int e = 0; e < EDIM; ++e) d += s_enc[e] * We[e * ADIM + a];
        float bev  = d / (1.0f + fabsf(d));                 // softsign
        float conv = conv_w[a * 31 + (15 - t)];             // conv of one-hot att0
        float ev   = bev + s_bspkr[a] + conv + bspd;
        s_red[a]   = tanhf(ev) * wpa;
        __syncthreads();
        for (int ofs = 128; ofs > 0; ofs >>= 1) {
            if (a < ofs) s_red[a] += s_red[a + ofs];
            __syncthreads();
        }
        if (a == 0) s_logit[t] = s_red[0] + b_proj[0];
        __syncthreads();
    }

    if (a == 0) {       // softmax over masked positions (global max cancels)
        float mx = -INFINITY;
        for (int t = 0; t <= tmax; ++t) mx = fmaxf(mx, s_logit[t]);
        float sum = 0.0f;
        for (int t = 0; t <= tmax; ++t) { float w = __expf(s_logit[t] - mx); s_att[t] = w; sum += w; }
        float inv = 1.0f / fmaxf(sum, 1e-12f);
        for (int t = 0; t <= tmax; ++t) s_att[t] *= inv;
    }
    __syncthreads();

    // context[e] = sum_t att[t] * (input_enc[n,t,e] + sp[n,e])
    const float* spv = sp_ws + n * EDIM;
    for (int e = a; e < EDIM; e += 256) {
        float c = 0.0f, sp = spv[e];
        for (int t = 0; t <= tmax; ++t)
            c += s_att[t] * (input_enc[((size_t)n * TENC + t) * EDIM + e] + sp);
        in_lstm[n * KIN0 + EDIM + e]  = c;
        dec_ws[n * DECD + H4 + e]     = c;
        out_ctx[n * EDIM + e]         = c;
        out_dec[n * DECD + H4 + e]    = c;
    }
    if (a < SDIM) in_lstm[n * KIN0 + EDIM + DDIM + a] = sv[a];
}

// ---------------------------------------------------------------------------
// fp32 WMMA GEMM: C(MxN) = A(MxK) * B + bias0 (+bias1)
//   WT=true : B = W^T, W stored row-major (N, K)   -> v2f loads on both sides
//   WT=false: B = W,   W stored row-major (K, N)
// One wave per 16 x (16*NT) tile; each A v2f load is reused by NT WMMAs.
// K must be a multiple of 16 (1088 / 2048 / 2560 all are).
// A layout: lane holds row M=lane&15, VGPR v holds K = v + 2*(lane>>4).
// B layout: lane holds col N=lane&15, VGPR v holds K = v + 2*(lane>>4).
// D layout: VGPR r -> row M = r + 8*(lane>>4), col N = lane&15.
// ---------------------------------------------------------------------------
template <bool WT, int NT>
__global__ __launch_bounds__(32) void gemm_wmma_f32(
    const float* __restrict__ A, int lda,
    const float* __restrict__ W, int ldw,
    const float* __restrict__ bias0, const float* __restrict__ bias1,
    float* __restrict__ C, int ldc, int K)
{
    const int lane   = threadIdx.x;
    const int nbase0 = blockIdx.x * 16 * NT;
    const int mbase  = blockIdx.y * 16;
    const int nl     = lane & 15;
    const int kh     = lane >> 4;       // 0 or 1

    v8f acc[NT];
    #pragma unroll
    for (int t = 0; t < NT; ++t) {
        const int nb = nbase0 + 16 * t + nl;
        float bv = bias0[nb] + (bias1 ? bias1[nb] : 0.0f);
        #pragma unroll
        for (int r = 0; r < 8; ++r) acc[t][r] = bv;  // same output col per lane
    }

    const float* arow = A + (size_t)(mbase + nl) * lda;
    const float* wrow[NT];
    #pragma unroll
    for (int t = 0; t < NT; ++t)
        wrow[t] = WT ? (W + (size_t)(nbase0 + 16 * t + nl) * ldw) : W;

    for (int kk = 0; kk < K; kk += 16) {
        if (WT) {
            #pragma unroll
            for (int t = 0; t < NT; ++t)                 // stream weights ahead
                __builtin_prefetch(wrow[t] + kk + 512, 0, 1);
        }
        #pragma unroll
        for (int u = 0; u < 4; ++u) {
            const int ka = kk + 4 * u + 2 * kh;
            v2f a = *(const v2f*)(arow + ka);            // A[m, ka..ka+1], 8B aligned
            #pragma unroll
            for (int t = 0; t < NT; ++t) {
                v2f b;
                if (WT) {
                    b = *(const v2f*)(wrow[t] + ka);     // W[n, ka..ka+1] == B[k, n]
                } else {
                    const int nb = nbase0 + 16 * t + nl;
                    b.x = W[(size_t)(ka + 0) * ldw + nb];
                    b.y = W[(size_t)(ka + 1) * ldw + nb];
                }
                acc[t] = __builtin_amdgcn_wmma_f32_16x16x4_f32(
                    /*neg_a=*/false, a, /*neg_b=*/false, b,
                    /*c_mod=*/(short)0, acc[t], /*reuse_a=*/false, /*reuse_b=*/false);
            }
        }
    }

    #pragma unroll
    for (int t = 0; t < NT; ++t)
        #pragma unroll
        for (int r = 0; r < 8; ++r)
            C[(size_t)(mbase + r + 8 * kh) * ldc + nbase0 + 16 * t + nl] = acc[t][r];
}

// ---------------------------------------------------------------------------
// LSTM pointwise with h=c=0:  c' = sig(i)*tanh(g);  h' = sig(o)*tanh(c')
// gates layout per row: [i(2048) f(2048) g(2048) o(2048)]
// ---------------------------------------------------------------------------
__global__ __launch_bounds__(256) void lstm_pointwise(
    const float* __restrict__ gates,
    float* __restrict__ out0, int stride0,
    float* __restrict__ out1, int stride1)
{
    const int idx = blockIdx.x * 256 + threadIdx.x;   // 64*2048 total
    const int n = idx / H4, j = idx % H4;
    const float* g = gates + (size_t)n * G4;
    float iv = g[j], gv = g[2 * H4 + j], ov = g[3 * H4 + j];
    float c  = sigmoidf(iv) * tanhf(gv);
    float h  = sigmoidf(ov) * tanhf(c);
    out0[(size_t)n * stride0 + j] = h;
    if (out1) out1[(size_t)n * stride1 + j] = h;
}

// ---------------------------------------------------------------------------
extern "C" void kernel_launch(void* const* d_in, const int* in_sizes, int n_in,
                              void* d_out, int out_size, void* d_ws, size_t ws_size,
                              hipStream_t stream) {
    const float* input_enc = (const float*)d_in[0];
    const float* input_dec = (const float*)d_in[1];
    const float* spkr_vec  = (const float*)d_in[2];
    const float* speed     = (const float*)d_in[3];
    const int*   lengths   = (const int*)  d_in[4];
    const float* We        = (const float*)d_in[5];
    const float* be        = (const float*)d_in[6];
    const float* Ws        = (const float*)d_in[7];
    const float* conv_w    = (const float*)d_in[8];
    const float* w_proj    = (const float*)d_in[9];
    const float* b_proj    = (const float*)d_in[10];
    const float* W_speed   = (const float*)d_in[11];
    const float* sp_W1     = (const float*)d_in[12];
    const float* sp_b1     = (const float*)d_in[13];
    const float* sp_W2     = (const float*)d_in[14];
    const float* sp_b2     = (const float*)d_in[15];
    const float* pre_W1    = (const float*)d_in[16];
    const float* pre_b1    = (const float*)d_in[17];
    const float* pre_W2    = (const float*)d_in[18];
    const float* pre_b2    = (const float*)d_in[19];
    const float* Wih0      = (const float*)d_in[20];
    /* Whh0 d_in[21] dead: h=0 */
    const float* bih0      = (const float*)d_in[22];
    const float* bhh0      = (const float*)d_in[23];
    const float* Wih1      = (const float*)d_in[24];
    /* Whh1 d_in[25] dead: h=0 */
    const float* bih1      = (const float*)d_in[26];
    const float* bhh1      = (const float*)d_in[27];
    const float* Wo        = (const float*)d_in[28];
    const float* bo        = (const float*)d_in[29];

    float* out_y   = (float*)d_out;                       // (64,2,80)  = 10240
    float* out_ctx = out_y + NB * NOUT;                   // (64,1,512) = 32768
    float* out_dec = out_ctx + NB * EDIM;                 // (64,1,2560)=163840

    float* ws = (float*)d_ws;
    float* ws_sp   = ws;                         // 64*512
    float* ws_inl  = ws_sp  + NB * EDIM;         // 64*1088
    float* ws_g0   = ws_inl + NB * KIN0;         // 64*8192
    float* ws_h1   = ws_g0  + NB * G4;           // 64*2048
    float* ws_g1   = ws_h1  + NB * H4;           // 64*8192
    float* ws_dec  = ws_g1  + NB * G4;           // 64*2560

    sp_kernel<<<NB, 256, 0, stream>>>(speed, sp_W1, sp_b1, sp_W2, sp_b2, ws_sp);
    prenet_kernel<<<NB, 256, 0, stream>>>(input_dec, spkr_vec, pre_W1, pre_b1,
                                          pre_W2, pre_b2, ws_inl);
    attn_kernel<<<NB, 256, 0, stream>>>(input_enc, spkr_vec, speed, lengths,
                                        We, be, Ws, conv_w, w_proj, b_proj,
                                        W_speed, ws_sp, ws_inl, ws_dec,
                                        out_ctx, out_dec);

    // gates0 = in_lstm(64x1088) @ Wih0^T(1088x8192) + bih0 + bhh0
    gemm_wmma_f32<true, 2><<<dim3(G4 / 32, NB / 16), 32, 0, stream>>>(
        ws_inl, KIN0, Wih0, KIN0, bih0, bhh0, ws_g0, G4, KIN0);
    lstm_pointwise<<<(NB * H4) / 256, 256, 0, stream>>>(ws_g0, ws_h1, H4,
                                                        nullptr, 0);

    // gates1 = h1(64x2048) @ Wih1^T(2048x8192) + bih1 + bhh1
    gemm_wmma_f32<true, 2><<<dim3(G4 / 32, NB / 16), 32, 0, stream>>>(
        ws_h1, H4, Wih1, H4, bih1, bhh1, ws_g1, G4, H4);
    // h2 -> dec_ws[:,0:2048] and d_out dec_output region
    lstm_pointwise<<<(NB * H4) / 256, 256, 0, stream>>>(ws_g1, ws_dec, DECD,
                                                        out_dec, DECD);

    // output = dec(64x2560) @ Wo(2560x160) + bo
    gemm_wmma_f32<false, 1><<<dim3(NOUT / 16, NB / 16), 32, 0, stream>>>(
        ws_dec, DECD, Wo, NOUT, bo, nullptr, out_y, NOUT, DECD);
}